// HadamardTransform_68891275428167
// MI455X (gfx1250) — compile-verified
//
#include <hip/hip_runtime.h>

// WHT-4096 over 16384 rows, single pass:
//   H_4096 = H16 (x) H16 (x) H16  (Sylvester, symmetric, entries +-1)
// Stage 1 (i0, contiguous digit): v_wmma_f32_16x16x4_f32 x4 chained per 16x16 tile.
// Stage 2 (i1): m-bits 0-2 live in accumulator VGPR index -> VALU butterflies;
//               m-bit 3 is the lane-half bit -> v_permlanex16 swap + fma.
// Stage 3 (i2): butterflies across the 16 register-resident tiles.
// One wave (32 lanes) per row; 8 waves per block; grid covers rows exactly
// so EXEC is all-ones (WMMA requirement). fp32 end to end.

typedef __attribute__((ext_vector_type(2))) float v2f;
typedef __attribute__((ext_vector_type(8))) float v8f;

__device__ __forceinline__ float swap_half16(float v) {
  // lane i <- lane (i ^ 16): identity selects into v_permlanex16
  int r = __builtin_amdgcn_permlanex16(0, __float_as_int(v),
                                       0x76543210, (int)0xfedcba98,
                                       false, false);
  return __int_as_float(r);
}

__global__ __launch_bounds__(256) void
HadamardTransform_68891275428167_kernel(const float* __restrict__ x,
                                        float* __restrict__ y) {
  const int lane = threadIdx.x & 31;
  const int wave = threadIdx.x >> 5;
  const size_t row = (size_t)blockIdx.x * 8 + wave;

  const int m  = lane & 15;   // A-operand M row / B-operand N col / C-D N col
  const int hi = lane >> 4;   // lane-half bit
  const float shalf = hi ? -1.0f : 1.0f;

  const float* __restrict__ xr = x + row * 4096;
  float*       __restrict__ yr = y + row * 4096;

  // H16 in B-operand layout (4 K-chunks of 4 rows; 2 VGPRs per chunk):
  //   v0 <-> K = 4c + 2*hi, v1 <-> K = 4c + 1 + 2*hi, N = lane%16
  v2f B[4];
#pragma unroll
  for (int c = 0; c < 4; ++c) {
    const int k0 = 4 * c + 2 * hi;
    B[c].x = (__popc(k0 & m) & 1) ? -1.0f : 1.0f;
    B[c].y = (__popc((k0 + 1) & m) & 1) ? -1.0f : 1.0f;
  }

  v8f d[16];

  // ---- Stage 1: contract i0 with WMMA.  Tile t = i2, A rows = i1, A cols = i0.
  // A-operand layout chunk c: lane holds (M=lane%16, K=4c+2*hi) and (.., K+1)
  // => one float2 load per chunk straight from HBM into operand registers.
#pragma unroll
  for (int t = 0; t < 16; ++t) {
    v8f acc = {0.f, 0.f, 0.f, 0.f, 0.f, 0.f, 0.f, 0.f};
#pragma unroll
    for (int c = 0; c < 4; ++c) {
      const v2f a = *(const v2f*)(xr + 256 * t + 16 * m + 4 * c + 2 * hi);
      acc = __builtin_amdgcn_wmma_f32_16x16x4_f32(
          /*neg_a=*/false, a, /*neg_b=*/false, B[c],
          /*c_mod=*/(short)0, acc, /*reuse_a=*/false, /*reuse_b=*/false);
    }
    d[t] = acc;
  }

  // ---- Stage 2: contract i1 (= m).  C/D layout: VGPR j holds rows j (lanes
  // 0-15) and j+8 (lanes 16-31).  Bits 0-2 of m -> VGPR butterflies, bit 3 ->
  // cross-half swap.
#pragma unroll
  for (int t = 0; t < 16; ++t) {
    float e[8];
#pragma unroll
    for (int j = 0; j < 8; ++j) e[j] = d[t][j];

#pragma unroll
    for (int b = 0; b < 3; ++b) {
      const int h = 1 << b;
#pragma unroll
      for (int g = 0; g < 8; ++g) {
        if (!(g & h)) {
          const float a0 = e[g], b0 = e[g + h];
          e[g]     = a0 + b0;
          e[g + h] = a0 - b0;
        }
      }
    }
    // m bit 3: partner is same VGPR, other lane half.
    //   new = other + shalf * mine  (hi=0: a+b, hi=1: a-b)
#pragma unroll
    for (int j = 0; j < 8; ++j) {
      const float u = swap_half16(e[j]);
      e[j] = fmaf(shalf, e[j], u);
    }
#pragma unroll
    for (int j = 0; j < 8; ++j) d[t][j] = e[j];
  }

  // ---- Stage 3: contract i2 (= tile index): butterflies across tiles.
#pragma unroll
  for (int b = 0; b < 4; ++b) {
    const int h = 1 << b;
#pragma unroll
    for (int t = 0; t < 16; ++t) {
      if (!(t & h)) {
        const v8f a0 = d[t], b0 = d[t + h];
        d[t]     = a0 + b0;
        d[t + h] = a0 - b0;
      }
    }
  }

  // ---- Store: element (t, M=j+8*hi, N=m) -> yr[256t + 16M + N]
#pragma unroll
  for (int t = 0; t < 16; ++t) {
#pragma unroll
    for (int j = 0; j < 8; ++j) {
      const int mm = j + 8 * hi;
      yr[256 * t + 16 * mm + m] = d[t][j];
    }
  }
}

extern "C" void kernel_launch(void* const* d_in, const int* in_sizes, int n_in,
                              void* d_out, int out_size, void* d_ws, size_t ws_size,
                              hipStream_t stream) {
  (void)n_in; (void)d_ws; (void)ws_size; (void)out_size;
  const float* x = (const float*)d_in[0];
  float* y = (float*)d_out;
  const int rows = in_sizes[0] / 4096;   // 16384
  const int blocks = rows / 8;           // 8 waves (rows) per 256-thread block
  HadamardTransform_68891275428167_kernel<<<blocks, 256, 0, stream>>>(x, y);
}